// LocalLoadBalancingLoss_2113123909879
// MI455X (gfx1250) — compile-verified
//
#include <hip/hip_runtime.h>

#define BATCH     65536
#define T_TOT     792
#define NDST      99
#define NLINK     16
#define ROWS_PB   128          // rows per block (8 waves x 16 rows)
#define KCH       88           // K-chunk (multiple of 4); 792 = 9 * 88
#define NCH       9
#define LSTRIDE   89           // odd stride -> conflict-free column reads
#define THREADS   256
#define NBLOCKS   (BATCH / ROWS_PB)   // 512

typedef float v2f __attribute__((ext_vector_type(2)));
typedef float v8f __attribute__((ext_vector_type(8)));

__global__ __launch_bounds__(THREADS)
void lbl_main_kernel(const float* __restrict__ pred,
                     const float* __restrict__ dem,
                     const int*   __restrict__ t2l,
                     const float* __restrict__ cap,
                     float*       __restrict__ partials)
{
    __shared__ float tt[ROWS_PB * LSTRIDE];   // 128 x 89 f32 = 45.6 KB
    __shared__ int   s_t2l[T_TOT];            // 3.1 KB
    __shared__ float s_red[(THREADS / 32) * 2];

    const int tid  = threadIdx.x;
    const int wave = tid >> 5;
    const int lane = tid & 31;
    const int half = lane >> 4;      // 0: lanes 0-15, 1: lanes 16-31
    const int mrow = lane & 15;      // row-in-tile (A/B), link id (C)
    const int row_base = blockIdx.x * ROWS_PB;
    const int r0 = wave * 16;        // this wave's 16-row tile

    for (int i = tid; i < T_TOT; i += THREADS) s_t2l[i] = t2l[i];

    const float inv_cap = 1.0f / (cap[mrow] + 1e-8f);

    v8f acc = {};   // C starts at zero

    for (int c = 0; c < NCH; ++c) {
        __syncthreads();
        // ---- stage chunk: tt[row][k] = pred * demand (coalesced in k) ----
        const int gk0 = c * KCH;
        for (int idx = tid; idx < ROWS_PB * KCH; idx += THREADS) {
            int row  = idx / KCH;
            int k    = idx - row * KCH;
            int gcol = gk0 + k;
            int grow = row_base + row;
            float p = pred[grow * T_TOT + gcol];
            float d = dem[grow * NDST + (gcol >> 3)];   // DEST_OF_TUNNEL[t] = t>>3
            tt[row * LSTRIDE + k] = p * d;
        }
        __syncthreads();
        // ---- WMMA chain: link_traffic += tunnel_traffic x one-hot(S) ----
        const float* arow = &tt[(r0 + mrow) * LSTRIDE];
        for (int kk = 0; kk < KCH; kk += 4) {
            // A fragment (16x4 f32): v0 = K=kk   (lo) / kk+2 (hi)
            //                        v1 = K=kk+1 (lo) / kk+3 (hi)
            int kA = kk + 2 * half;
            v2f a;
            a.x = arow[kA];
            a.y = arow[kA + 1];
            // B fragment (4x16 one-hot): same K split, N = mrow
            int l0 = s_t2l[gk0 + kA];
            int l1 = s_t2l[gk0 + kA + 1];
            v2f b;
            b.x = (l0 == mrow) ? 1.0f : 0.0f;
            b.y = (l1 == mrow) ? 1.0f : 0.0f;
            acc = __builtin_amdgcn_wmma_f32_16x16x4_f32(
                false, a, false, b, (short)0, acc, false, false);
        }
    }

    // acc[j] = link_traffic[row = r0 + j + 8*half][link = mrow]
    float local = 0.0f;
#pragma unroll
    for (int j = 0; j < 8; ++j) {
        float u  = acc[j] * inv_cap;
        float s  = u;
        float ss = u * u;
        float mx = u;
#pragma unroll
        for (int m = 8; m >= 1; m >>= 1) {   // reduce within each 16-lane half
            s  += __shfl_xor(s,  m, 32);
            ss += __shfl_xor(ss, m, 32);
            float o = __shfl_xor(mx, m, 32);
            mx = (mx > o) ? mx : o;
        }
        float var = (ss - s * s * (1.0f / 16.0f)) * (1.0f / 15.0f);  // ddof=1
        local += var + 0.5f * mx;
    }

    __syncthreads();                         // done with tt before reuse of LDS phase
    if (mrow == 0) s_red[wave * 2 + half] = local;   // lanes 0 & 16 hold 8 rows each
    __syncthreads();
    if (tid == 0) {
        float sum = 0.0f;
#pragma unroll
        for (int i = 0; i < (THREADS / 32) * 2; ++i) sum += s_red[i];
        partials[blockIdx.x] = sum;
    }
}

__global__ __launch_bounds__(256)
void lbl_reduce_kernel(const float* __restrict__ partials,
                       float* __restrict__ out, int n)
{
    __shared__ float s[256];
    float v = 0.0f;
    for (int i = threadIdx.x; i < n; i += 256) v += partials[i];
    s[threadIdx.x] = v;
    __syncthreads();
    for (int off = 128; off >= 1; off >>= 1) {
        if ((int)threadIdx.x < off) s[threadIdx.x] += s[threadIdx.x + off];
        __syncthreads();
    }
    if (threadIdx.x == 0) out[0] = s[0] * (1.0f / (float)BATCH);
}

extern "C" void kernel_launch(void* const* d_in, const int* in_sizes, int n_in,
                              void* d_out, int out_size, void* d_ws, size_t ws_size,
                              hipStream_t stream)
{
    const float* pred = (const float*)d_in[0];   // [65536, 792]
    const float* dem  = (const float*)d_in[1];   // [65536, 99]
    const int*   t2l  = (const int*)  d_in[2];   // [792]
    const float* cap  = (const float*)d_in[3];   // [16]
    float* out      = (float*)d_out;
    float* partials = (float*)d_ws;              // 512 floats = 2 KB scratch

    lbl_main_kernel<<<NBLOCKS, THREADS, 0, stream>>>(pred, dem, t2l, cap, partials);
    lbl_reduce_kernel<<<1, 256, 0, stream>>>(partials, out, NBLOCKS);
}